// HeterophilyLinearAgg_39273180955310
// MI455X (gfx1250) — compile-verified
//
#include <hip/hip_runtime.h>
#include <hip/hip_bf16.h>
#include <math.h>

typedef __attribute__((ext_vector_type(2))) float v2f;
typedef __attribute__((ext_vector_type(8))) float v8f;

#define D 128          // feature dim (in == out == 128)
#define LN_EPS 1e-5f

// ---------------------------------------------------------------------------
// CDNA5 async memory->LDS copy (16B per lane), tracked with ASYNCcnt.
// LDS destination byte address = low 32 bits of the generic pointer
// (flat LDS addresses carry the LDS offset in addr[31:0]).
// ---------------------------------------------------------------------------
__device__ __forceinline__ void async_copy_b128(void* lds_dst, const void* gsrc) {
    unsigned loff = (unsigned)(uintptr_t)lds_dst;
    asm volatile("global_load_async_to_lds_b128 %0, %1, off"
                 :: "v"(loff), "v"(gsrc) : "memory");
}
__device__ __forceinline__ void wait_async_all() {
    asm volatile("s_wait_asynccnt 0" ::: "memory");
}

// ---------------------------------------------------------------------------
// Fold the 4 weight matrices + sigmoid branch scales into 3 combined K-major
// matrices:  Wc[k*128+n]  so the WMMA B-fragment loads are coalesced.
//   z = h @ WcA + nb1 @ WcB + nb2 @ WcC + bias
// ---------------------------------------------------------------------------
__global__ void combine_weights_kernel(const float* __restrict__ Wself,
                                       const float* __restrict__ Wnb1,
                                       const float* __restrict__ Whp,
                                       const float* __restrict__ Wnb2,
                                       const float* __restrict__ logits,
                                       float* __restrict__ WcA,
                                       float* __restrict__ WcB,
                                       float* __restrict__ WcC) {
    int idx = blockIdx.x * blockDim.x + threadIdx.x;   // exactly 128*128 threads
    int k = idx >> 7;
    int n = idx & 127;
    float s0 = 2.0f / (1.0f + expf(-logits[0]));
    float s1 = 2.0f / (1.0f + expf(-logits[1]));
    float s2 = 2.0f / (1.0f + expf(-logits[2]));
    float s3 = 2.0f / (1.0f + expf(-logits[3]));
    float wS = Wself[n * D + k];
    float w1 = Wnb1[n * D + k];
    float wH = Whp[n * D + k];
    float w2 = Wnb2[n * D + k];
    WcA[idx] = s0 * wS + s2 * wH;   // multiplies h   (self + high-pass)
    WcB[idx] = s1 * w1 - s2 * wH;   // multiplies nb1 (nb1 - high-pass part)
    WcC[idx] = s3 * w2;             // multiplies nb2
}

// ---------------------------------------------------------------------------
// Fused scatter-MEAN: one wave32 per edge. Lane l loads float4 l of the 512B
// source row (one coalesced burst), scales by 1/deg[dst] (mean folded into
// the scatter: sum(x/deg) == sum(x)/deg), and does 4 f32 atomics into the
// contiguous destination row. Prefetch a future edge's source row into L2.
// ---------------------------------------------------------------------------
__global__ void scatter_mean_kernel(const float4* __restrict__ x,   // [N][32]
                                    const int* __restrict__ src,
                                    const int* __restrict__ dst,
                                    const float* __restrict__ deg,  // [N]
                                    float* __restrict__ acc,        // [N][128]
                                    int E) {
    int lane = threadIdx.x & 31;
    int e = blockIdx.x * (blockDim.x >> 5) + (threadIdx.x >> 5);
    if (e >= E) return;
    if (e + 8 < E) {
        const void* p = (const void*)(x + (size_t)src[e + 8] * 32 + lane);
        __builtin_prefetch(p, 0, 1);   // global_prefetch_b8
    }
    int s = src[e];
    int d = dst[e];
    float invd = 1.0f / deg[d];        // broadcast load, one per wave
    float4 v = x[(size_t)s * 32 + lane];
    float* o = acc + (size_t)d * D + lane * 4;
    atomicAdd(o + 0, v.x * invd);
    atomicAdd(o + 1, v.y * invd);
    atomicAdd(o + 2, v.z * invd);
    atomicAdd(o + 3, v.w * invd);
}

// ---------------------------------------------------------------------------
// Fused triple-GEMM (f32 WMMA 16x16x4) + bias + LayerNorm.
// One block = 16 node rows. 8 waves, wave w owns output columns [16w,16w+16).
// A tiles (h / nb1 / nb2, 16x128 each) staged into LDS via
// global_load_async_to_lds_b128 with a 132-float row pitch: the v2f fragment
// read pattern (lane<16: K=4k..4k+1 of row lane; lane>=16: K=4k+2..4k+3)
// then touches all 64 LDS banks exactly once.
// ---------------------------------------------------------------------------
__global__ __launch_bounds__(256)
void gemm_ln_kernel(const float* __restrict__ h,
                    const float* __restrict__ nb1,
                    const float* __restrict__ nb2,
                    const float* __restrict__ WcA,
                    const float* __restrict__ WcB,
                    const float* __restrict__ WcC,
                    const float* __restrict__ bias,
                    const float* __restrict__ gamma,
                    const float* __restrict__ beta,
                    float* __restrict__ out, int N) {
    __shared__ __align__(16) float At[3][16][132];
    __shared__ float Zt[16][132];

    const int tid  = threadIdx.x;
    const int row0 = blockIdx.x * 16;

    // ---- async, coalesced staging of the three 16x128 A tiles ----
    const float* Xs[3] = {h, nb1, nb2};
    for (int m = 0; m < 3; ++m) {
        const float* X = Xs[m];
        for (int i = tid; i < 16 * 32; i += 256) {
            int r  = i >> 5;
            int c4 = i & 31;
            int gr = row0 + r;
            if (gr >= N) gr = N - 1;                 // N==100000 is 16-aligned; safety
            async_copy_b128(&At[m][r][c4 * 4], X + (size_t)gr * D + c4 * 4);
        }
    }
    wait_async_all();          // this wave's async LDS writes have landed
    __syncthreads();           // all waves' writes visible

    const int wave  = tid >> 5;        // column tile 0..7
    const int lane  = tid & 31;
    const int lrow  = lane & 15;
    const int khalf = lane >> 4;       // 0: K pair 0-1, 1: K pair 2-3
    const int ncol  = wave * 16 + lrow;

    const float* Ws[3] = {WcA, WcB, WcC};
    v8f c = {};
    for (int m = 0; m < 3; ++m) {
        const float* W = Ws[m];
#pragma unroll 4
        for (int k = 0; k < 32; ++k) {
            int kb = 4 * k + 2 * khalf;
            v2f a, b;
            a.x = At[m][lrow][kb + 0];
            a.y = At[m][lrow][kb + 1];
            b.x = W[kb * D + ncol];                  // coalesced across lrow
            b.y = W[(kb + 1) * D + ncol];
            c = __builtin_amdgcn_wmma_f32_16x16x4_f32(
                    /*neg_a=*/false, a, /*neg_b=*/false, b,
                    /*c_mod=*/(short)0, c, /*reuse_a=*/false, /*reuse_b=*/false);
        }
    }

    // ---- spill z tile (+bias) to LDS for the cross-tile LayerNorm ----
    float bval = bias[ncol];
#pragma unroll
    for (int r = 0; r < 8; ++r) {
        // D layout: lanes 0-15 rows 0-7, lanes 16-31 rows 8-15
        Zt[r + 8 * khalf][ncol] = c[r] + bval;
    }
    __syncthreads();

    // ---- LayerNorm: 16 lanes per row, 8 columns per lane ----
    int row = tid >> 4;
    int sub = tid & 15;
    float s = 0.0f, ss = 0.0f;
#pragma unroll
    for (int j = 0; j < 8; ++j) {
        float z = Zt[row][sub + 16 * j];
        s  += z;
        ss += z * z;
    }
#pragma unroll
    for (int off = 8; off; off >>= 1) {
        s  += __shfl_xor(s,  off, 16);
        ss += __shfl_xor(ss, off, 16);
    }
    float mean = s * (1.0f / 128.0f);
    float var  = ss * (1.0f / 128.0f) - mean * mean;
    float rstd = rsqrtf(var + LN_EPS);

    int grow = row0 + row;
    if (grow < N) {
#pragma unroll
        for (int j = 0; j < 8; ++j) {
            int colj = sub + 16 * j;
            float z = Zt[row][colj];
            out[(size_t)grow * D + colj] = (z - mean) * rstd * gamma[colj] + beta[colj];
        }
    }
}

// ---------------------------------------------------------------------------
extern "C" void kernel_launch(void* const* d_in, const int* in_sizes, int n_in,
                              void* d_out, int out_size, void* d_ws, size_t ws_size,
                              hipStream_t stream) {
    const float* h      = (const float*)d_in[0];
    const int*   ei     = (const int*)d_in[1];   // [2][E] (src row, dst row)
    const float* deg    = (const float*)d_in[2];
    const float* Wself  = (const float*)d_in[3];
    const float* Wnb1   = (const float*)d_in[4];
    const float* Whp    = (const float*)d_in[5];
    const float* Wnb2   = (const float*)d_in[6];
    const float* bias   = (const float*)d_in[7];
    const float* logits = (const float*)d_in[8];
    const float* gamma  = (const float*)d_in[9];
    const float* beta   = (const float*)d_in[10];

    const int N = in_sizes[0] / D;
    const int E = in_sizes[1] / 2;

    // workspace layout: nb1 | nb2 | WcA | WcB | WcC  (~102.6 MB)
    float* nb1 = (float*)d_ws;
    float* nb2 = nb1 + (size_t)N * D;
    float* WcA = nb2 + (size_t)N * D;
    float* WcB = WcA + D * D;
    float* WcC = WcB + D * D;

    (void)hipMemsetAsync(nb1, 0, (size_t)N * D * sizeof(float), stream);
    (void)hipMemsetAsync(nb2, 0, (size_t)N * D * sizeof(float), stream);

    combine_weights_kernel<<<64, 256, 0, stream>>>(Wself, Wnb1, Whp, Wnb2, logits,
                                                   WcA, WcB, WcC);

    const int* srcI = ei;
    const int* dstI = ei + E;
    int sblocks = (E + 7) / 8;                 // 8 waves (edges) per 256-thread block
    // pass 1: nb1 = mean_agg(h)
    scatter_mean_kernel<<<sblocks, 256, 0, stream>>>((const float4*)h, srcI, dstI,
                                                     deg, nb1, E);
    // pass 2: nb2 = mean_agg(nb1)
    scatter_mean_kernel<<<sblocks, 256, 0, stream>>>((const float4*)nb1, srcI, dstI,
                                                     deg, nb2, E);

    gemm_ln_kernel<<<(N + 15) / 16, 256, 0, stream>>>(h, nb1, nb2, WcA, WcB, WcC,
                                                      bias, gamma, beta,
                                                      (float*)d_out, N);
}